// HGTLayerModified_18691697672935
// MI455X (gfx1250) — compile-verified
//
#include <hip/hip_runtime.h>
#include <math.h>

#define DIM 256
#define NHEAD 8
#define DKH 32

typedef float v2f __attribute__((ext_vector_type(2)));
typedef float v8f __attribute__((ext_vector_type(8)));

// ---------- monotonic float <-> uint key for atomic-max softmax ----------
__device__ __forceinline__ unsigned fkey(float x) {
  unsigned b = __float_as_uint(x);
  return (b & 0x80000000u) ? ~b : (b | 0x80000000u);
}
__device__ __forceinline__ float finv(unsigned u) {
  return __uint_as_float((u & 0x80000000u) ? (u & 0x7fffffffu) : ~u);
}

// =====================================================================
// WMMA fp32 GEMM: C[M,256] = A[M,256] @ B[256,256] (+bias)
// optional epilogue: C = C*sigmoid(skip_param) + skip_feat*(1-sigmoid)
// Each wave computes a 64x16 tile (4 row-tiles, 4 accumulators) so the
// strided B fragment is amortized over 4 WMMAs.
// grid = (ceil(M/64), 2), block = 256 (8 waves -> 8 column tiles)
// =====================================================================
__global__ void gemm256_wmma(const float* __restrict__ A,
                             const float* __restrict__ B,
                             const float* __restrict__ bias,
                             const float* __restrict__ skip_feat,
                             const float* __restrict__ skip_param,
                             float* __restrict__ C, int M) {
  const int lane = threadIdx.x & 31;
  const int wave = threadIdx.x >> 5;
  const int row0 = blockIdx.x * 64;
  const int n0   = (blockIdx.y * 8 + wave) * 16;
  const int half = lane >> 4;   // 0: K=ka,ka+1   1: K=ka+2,ka+3
  const int l16  = lane & 15;
  const int nb   = n0 + l16;

  const float* Arow[4];
#pragma unroll
  for (int i = 0; i < 4; ++i) {
    int r = row0 + i * 16 + l16;
    if (r >= M) r = M - 1;      // clamp: keep EXEC all-ones for WMMA
    Arow[i] = A + (size_t)r * DIM;
  }

  v8f acc[4];
#pragma unroll
  for (int i = 0; i < 4; ++i) acc[i] = (v8f){0.f,0.f,0.f,0.f,0.f,0.f,0.f,0.f};

  for (int k = 0; k < DIM; k += 4) {
    const int ka = k + half * 2;
    // B-frag 4x16: VGPR0 = row ka (cols on lanes), VGPR1 = row ka+1
    v2f b;
    b.x = B[(size_t)ka * DIM + nb];
    b.y = B[(size_t)(ka + 1) * DIM + nb];
    // A-frags 16x4: lane row = l16; VGPR0 = K=ka, VGPR1 = K=ka+1
    v2f a0 = *reinterpret_cast<const v2f*>(Arow[0] + ka);
    v2f a1 = *reinterpret_cast<const v2f*>(Arow[1] + ka);
    v2f a2 = *reinterpret_cast<const v2f*>(Arow[2] + ka);
    v2f a3 = *reinterpret_cast<const v2f*>(Arow[3] + ka);
    acc[0] = __builtin_amdgcn_wmma_f32_16x16x4_f32(false, a0, false, b, (short)0, acc[0], false, false);
    acc[1] = __builtin_amdgcn_wmma_f32_16x16x4_f32(false, a1, false, b, (short)0, acc[1], false, false);
    acc[2] = __builtin_amdgcn_wmma_f32_16x16x4_f32(false, a2, false, b, (short)0, acc[2], false, false);
    acc[3] = __builtin_amdgcn_wmma_f32_16x16x4_f32(false, a3, false, b, (short)0, acc[3], false, false);
  }

  const float bv = bias ? bias[nb] : 0.f;
  float alpha = 0.f;
  if (skip_feat) alpha = 1.f / (1.f + __expf(-skip_param[0]));
#pragma unroll
  for (int i = 0; i < 4; ++i) {
#pragma unroll
    for (int j = 0; j < 8; ++j) {
      const int m = row0 + i * 16 + j + half * 8;  // C layout: VGPRj -> M=j (+8 hi)
      if (m < M) {
        float v = acc[i][j] + bv;
        if (skip_feat)
          v = v * alpha + skip_feat[(size_t)m * DIM + nb] * (1.f - alpha);
        C[(size_t)m * DIM + nb] = v;
      }
    }
  }
}

// =====================================================================
// Scatter per-head att/msg (H,32,32) into block-diagonal 256x256
// =====================================================================
__global__ void build_bd(const float* __restrict__ att,
                         const float* __restrict__ msg,
                         float* __restrict__ attbd,
                         float* __restrict__ msgbd) {
  const int idx = blockIdx.x * blockDim.x + threadIdx.x;  // 65536 threads
  if (idx >= DIM * DIM) return;
  const int r = idx >> 8, c = idx & 255;
  const int h = r >> 5;
  float va = 0.f, vm = 0.f;
  if ((c >> 5) == h) {
    const int o = h * (DKH * DKH) + (r & 31) * DKH + (c & 31);
    va = att[o];
    vm = msg[o];
  }
  attbd[idx] = va;
  msgbd[idx] = vm;
}

// =====================================================================
// Pass 1: att_e[e,h] = <q[dst], k_rel[src]>_h * pri[h]/sqrt(DK); seg atomic max
// one wave per edge; lane -> head = lane>>2, 8 elems each
// =====================================================================
__global__ void edge_att(const int* __restrict__ src, const int* __restrict__ dst,
                         const float* __restrict__ q, const float* __restrict__ krel,
                         const float* __restrict__ pri,
                         float* __restrict__ att_e, unsigned* __restrict__ segmax,
                         int E, float inv_sqrt_dk) {
  const int e = (blockIdx.x * blockDim.x + threadIdx.x) >> 5;
  if (e >= E) return;
  const int lane = threadIdx.x & 31;
  const int h = lane >> 2;
  const int d0 = (lane & 3) * 8;
  const int s = src[e], t = dst[e];
  const float* qp = q + (size_t)t * DIM + h * DKH + d0;
  const float* kp = krel + (size_t)s * DIM + h * DKH + d0;
  float sum = 0.f;
#pragma unroll
  for (int j = 0; j < 8; ++j) sum += qp[j] * kp[j];
  sum += __shfl_xor(sum, 1);
  sum += __shfl_xor(sum, 2);
  if ((lane & 3) == 0) {
    const float a = sum * pri[h] * inv_sqrt_dk;
    att_e[(size_t)e * NHEAD + h] = a;
    atomicMax(&segmax[(size_t)t * NHEAD + h], fkey(a));
  }
}

// =====================================================================
// Pass 2: e = exp(a - max[dst]); seg atomic sum. one thread per (e,h)
// =====================================================================
__global__ void edge_exp(const int* __restrict__ dst, float* __restrict__ att_e,
                         const unsigned* __restrict__ segmax,
                         float* __restrict__ segsum, int E) {
  const int idx = blockIdx.x * blockDim.x + threadIdx.x;
  if (idx >= E * NHEAD) return;
  const int e = idx >> 3, h = idx & 7;
  const int t = dst[e];
  const float m = finv(segmax[(size_t)t * NHEAD + h]);
  const float v = __expf(att_e[idx] - m);
  att_e[idx] = v;
  atomicAdd(&segsum[(size_t)t * NHEAD + h], v);
}

// =====================================================================
// Pass 3: hacc[dst] += (e/sum) * v_rel[src]. one wave per edge
// =====================================================================
__global__ void edge_agg(const int* __restrict__ src, const int* __restrict__ dst,
                         const float* __restrict__ att_e,
                         const float* __restrict__ segsum,
                         const float* __restrict__ vrel,
                         float* __restrict__ hacc, int E) {
  const int e = (blockIdx.x * blockDim.x + threadIdx.x) >> 5;
  if (e >= E) return;
  const int lane = threadIdx.x & 31;
  const int h = lane >> 2;
  const int d0 = (lane & 3) * 8;
  const int s = src[e], t = dst[e];
  const float w = att_e[(size_t)e * NHEAD + h] / segsum[(size_t)t * NHEAD + h];
  const float* vp = vrel + (size_t)s * DIM + h * DKH + d0;
  float* hp = hacc + (size_t)t * DIM + h * DKH + d0;
#pragma unroll
  for (int j = 0; j < 8; ++j) atomicAdd(&hp[j], w * vp[j]);
}

// =====================================================================
// t += weight * relu(hacc)
// =====================================================================
__global__ void relu_acc(const float* __restrict__ hacc, float* __restrict__ tbuf,
                         float w, int n) {
  const int idx = blockIdx.x * blockDim.x + threadIdx.x;
  if (idx < n) tbuf[idx] += w * fmaxf(hacc[idx], 0.f);
}

// =====================================================================
extern "C" void kernel_launch(void* const* d_in, const int* in_sizes, int n_in,
                              void* d_out, int out_size, void* d_ws, size_t ws_size,
                              hipStream_t stream) {
  (void)n_in; (void)ws_size;
  // ---- inputs (setup_inputs order) ----
  const float* feats[3] = {(const float*)d_in[0],   // word
                           (const float*)d_in[1],   // topic
                           (const float*)d_in[2]};  // doc
  const int NW = in_sizes[0] / DIM, NT = in_sizes[1] / DIM, ND = in_sizes[2] / DIM;
  const int nN[3] = {NW, NT, ND};
  // edges: ww, wt, wd, tt, td
  const int* esrc[5] = {(const int*)d_in[3], (const int*)d_in[5], (const int*)d_in[7],
                        (const int*)d_in[9], (const int*)d_in[11]};
  const int* edst[5] = {(const int*)d_in[4], (const int*)d_in[6], (const int*)d_in[8],
                        (const int*)d_in[10], (const int*)d_in[12]};
  int Ecnt[5];
  for (int i = 0; i < 5; ++i) Ecnt[i] = in_sizes[3 + 2 * i];

  // ---- params: jax pytree flatten, alphabetical dict keys ----
  // top: a, att, k, msg, pri, q, skip, v ; types sorted: doc,topic,word ;
  // rels sorted: td,tt,wd,wt,ww ; leaf order within type: W,b
  const float* aW[3]  = {(const float*)d_in[17], (const float*)d_in[15], (const float*)d_in[13]}; // word,topic,doc
  const float* ab[3]  = {(const float*)d_in[18], (const float*)d_in[16], (const float*)d_in[14]};
  // rel order here: ww, wt, wd, tt, td
  const float* attP[5] = {(const float*)d_in[23], (const float*)d_in[22], (const float*)d_in[21],
                          (const float*)d_in[20], (const float*)d_in[19]};
  const float* kW[2]  = {(const float*)d_in[28], (const float*)d_in[26]};  // word, topic
  const float* kb[2]  = {(const float*)d_in[29], (const float*)d_in[27]};
  const float* msgP[5] = {(const float*)d_in[34], (const float*)d_in[33], (const float*)d_in[32],
                          (const float*)d_in[31], (const float*)d_in[30]};
  const float* priP[5] = {(const float*)d_in[39], (const float*)d_in[38], (const float*)d_in[37],
                          (const float*)d_in[36], (const float*)d_in[35]};
  const float* qW[3]  = {(const float*)d_in[44], (const float*)d_in[42], (const float*)d_in[40]};
  const float* qb[3]  = {(const float*)d_in[45], (const float*)d_in[43], (const float*)d_in[41]};
  const float* skipP[3] = {(const float*)d_in[48], (const float*)d_in[47], (const float*)d_in[46]};
  const float* vW[2]  = {(const float*)d_in[53], (const float*)d_in[51]};
  const float* vb[2]  = {(const float*)d_in[54], (const float*)d_in[52]};

  // ---- workspace carve ----
  size_t off = 0;
  auto alloc = [&](size_t floats) -> float* {
    float* p = (float*)((char*)d_ws + off);
    off += floats * sizeof(float);
    off = (off + 255) & ~size_t(255);
    return p;
  };
  float* qbuf[3]; for (int t = 0; t < 3; ++t) qbuf[t] = alloc((size_t)nN[t] * DIM);
  float* kbuf[2]; for (int t = 0; t < 2; ++t) kbuf[t] = alloc((size_t)nN[t] * DIM);
  float* vbuf[2]; for (int t = 0; t < 2; ++t) vbuf[t] = alloc((size_t)nN[t] * DIM);
  float* attbd = alloc(DIM * DIM);
  float* msgbd = alloc(DIM * DIM);
  float* krel  = alloc((size_t)NW * DIM);
  float* vrel  = alloc((size_t)NW * DIM);
  int Emax = 0; for (int i = 0; i < 5; ++i) Emax = Ecnt[i] > Emax ? Ecnt[i] : Emax;
  float* att_e = alloc((size_t)Emax * NHEAD);
  unsigned* segmax = (unsigned*)alloc((size_t)NW * NHEAD);
  float* segsum = alloc((size_t)NW * NHEAD);
  float* hacc = alloc((size_t)NW * DIM);
  float* tbuf[3]; for (int t = 0; t < 3; ++t) tbuf[t] = alloc((size_t)nN[t] * DIM);

  const float inv_sqrt_dk = 1.f / sqrtf((float)DKH);
  const dim3 blk(256);

  auto gemm = [&](const float* A, const float* B, const float* bias,
                  const float* sf, const float* sp, float* C, int M) {
    dim3 g((M + 63) / 64, 2);
    hipLaunchKernelGGL(gemm256_wmma, g, blk, 0, stream, A, B, bias, sf, sp, C, M);
  };

  // ---- phase 1: projections (WMMA GEMMs) ----
  for (int t = 0; t < 3; ++t) gemm(feats[t], qW[t], qb[t], nullptr, nullptr, qbuf[t], nN[t]);
  for (int t = 0; t < 2; ++t) gemm(feats[t], kW[t], kb[t], nullptr, nullptr, kbuf[t], nN[t]);
  for (int t = 0; t < 2; ++t) gemm(feats[t], vW[t], vb[t], nullptr, nullptr, vbuf[t], nN[t]);

  // zero t accumulators (contiguous: each size is a multiple of 1KB, no gaps)
  hipMemsetAsync(tbuf[0], 0, (size_t)(NW + NT + ND) * DIM * sizeof(float), stream);

  // ---- phase 2: per-relation attention aggregation ----
  // rel: ww, wt, wd, tt, td
  const int relSrcT[5] = {0, 0, 0, 1, 1};   // word,word,word,topic,topic
  const int relDstT[5] = {0, 1, 2, 1, 2};   // word,topic,doc,topic,doc
  const float relW[5]  = {1.0f, 0.5f, 0.5f, 0.5f, 0.5f};

  for (int r = 0; r < 5; ++r) {
    const int st = relSrcT[r], dt = relDstT[r];
    const int nS = nN[st], nD = nN[dt], E = Ecnt[r];

    hipLaunchKernelGGL(build_bd, dim3((DIM * DIM + 255) / 256), blk, 0, stream,
                       attP[r], msgP[r], attbd, msgbd);
    gemm(kbuf[st], attbd, nullptr, nullptr, nullptr, krel, nS);
    gemm(vbuf[st], msgbd, nullptr, nullptr, nullptr, vrel, nS);

    hipMemsetAsync(segmax, 0, (size_t)nD * NHEAD * sizeof(unsigned), stream);
    hipMemsetAsync(segsum, 0, (size_t)nD * NHEAD * sizeof(float), stream);
    hipMemsetAsync(hacc, 0, (size_t)nD * DIM * sizeof(float), stream);

    hipLaunchKernelGGL(edge_att, dim3((E + 7) / 8), blk, 0, stream,
                       esrc[r], edst[r], qbuf[dt], krel, priP[r],
                       att_e, segmax, E, inv_sqrt_dk);
    hipLaunchKernelGGL(edge_exp, dim3((E * NHEAD + 255) / 256), blk, 0, stream,
                       edst[r], att_e, segmax, segsum, E);
    hipLaunchKernelGGL(edge_agg, dim3((E + 7) / 8), blk, 0, stream,
                       esrc[r], edst[r], att_e, segsum, vrel, hacc, E);
    hipLaunchKernelGGL(relu_acc, dim3((nD * DIM + 255) / 256), blk, 0, stream,
                       hacc, tbuf[dt], relW[r], nD * DIM);
  }

  // ---- phase 3: output projection + sigmoid-skip residual ----
  float* outp = (float*)d_out;
  float* outs[3] = {outp, outp + (size_t)NW * DIM, outp + (size_t)(NW + NT) * DIM};
  for (int t = 0; t < 3; ++t)
    gemm(tbuf[t], aW[t], ab[t], feats[t], skipP[t], outs[t], nN[t]);
  (void)out_size;
}